// Canny_1374389534866
// MI455X (gfx1250) — compile-verified
//
#include <hip/hip_runtime.h>
#include <math.h>

// ---------------------------------------------------------------------------
// Canny pipeline for MI455X (gfx1250, wave32).
// Kernel 1: fused gray + 7x7 Gaussian (via V_WMMA_F32_16X16X4_F32) + Sobel +
//           magnitude + orientation(img0) + per-image LDS histogram.
// Kernel 2: 85% quantile selection from histogram.
// Kernel 3: streaming non-max suppression + threshold.
// ---------------------------------------------------------------------------

typedef float v2f __attribute__((ext_vector_type(2)));
typedef float v8f __attribute__((ext_vector_type(8)));

#define TILE     64
#define GRAY_W   88     // 72 valid cols + pad for WMMA over-read
#define GRAY_H   80     // 72 valid rows + pad
#define HOUT_W   80
#define HOUT_H   88     // v-pass reads rows up to 87 (pad rows zeroed)
#define SM_W     80
#define NBINS    2048
#define HIST_MAX 16.0f  // histogram value range [0, 16)

// Gaussian g1 (sigma=0.8, 7 taps, normalized), symmetric.
__device__ __forceinline__ float gcoef(int d) {
    // d in [0..6]; returns g1[d]; 0 outside. Branchless select chain (no
    // dynamic array indexing -> no scratch).
    float w = 0.0f;
    w = (d == 0 || d == 6) ? 0.000440744f : w;
    w = (d == 1 || d == 5) ? 0.021910500f : w;
    w = (d == 2 || d == 4) ? 0.228311000f : w;
    w = (d == 3)           ? 0.498676000f : w;
    return w;
}
// Band matrix entry W[k][n] = g1[k-n] (used for both passes).
__device__ __forceinline__ float bandw(int k, int n) { return gcoef(k - n); }

__device__ __forceinline__ int reflect512(int t) {
    t = (t < 0) ? -t : t;
    t = (t > 511) ? (1022 - t) : t;
    return t;
}

__global__ __launch_bounds__(256) void k_grad(const float* __restrict__ x,
                                              float* __restrict__ mag,
                                              unsigned char* __restrict__ idxMap,
                                              unsigned* __restrict__ gHist) {
    __shared__ __align__(16) union {
        float gray[GRAY_H * GRAY_W];   // stage A input (channel mean)
        float smooth[GRAY_H * SM_W];   // stage B2 result (aliased: gray dead)
    } uA;
    __shared__ __align__(16) float hout[HOUT_H * HOUT_W];
    __shared__ unsigned hist[NBINS];

    const int tid  = threadIdx.x;
    const int img  = blockIdx.z;
    const int ty   = blockIdx.y, tx = blockIdx.x;
    const int baseY = ty * TILE - 4;
    const int baseX = tx * TILE - 4;

    // Zero LDS (pads must be 0 so WMMA over-read tiles stay finite).
    for (int i = tid; i < GRAY_H * GRAY_W; i += 256) uA.gray[i] = 0.0f;
    for (int i = tid; i < HOUT_H * HOUT_W; i += 256) hout[i] = 0.0f;
    for (int i = tid; i < NBINS; i += 256) hist[i] = 0u;
    __syncthreads();

    // ---- Stage A: grayscale (channel mean) with reflect padding ----------
    const size_t plane = 512u * 512u;
    const float* xb = x + (size_t)img * 3u * plane;
    const float inv3 = 1.0f / 3.0f;
    for (int i = tid; i < 72 * 72; i += 256) {
        int r = i / 72, c = i % 72;
        int yy = reflect512(baseY + r);
        int xx = reflect512(baseX + c);
        size_t o = (size_t)yy * 512u + (size_t)xx;
        uA.gray[r * GRAY_W + c] = (xb[o] + xb[plane + o] + xb[2u * plane + o]) * inv3;
    }
    __syncthreads();

    const int wave = tid >> 5;
    const int lane = tid & 31;
    const int m16  = lane & 15;     // row (A) / col (B,D) within 16x16
    const int half = lane >> 4;     // lane half selects K pair / D row half

    // ---- Stage B1: horizontal Gaussian via WMMA f32 16x16x4 --------------
    // hout[r][c] = sum_j g1[j] * gray[r][c+j]   (A = gray data, B = band)
    for (int t = wave; t < 25; t += 8) {
        int tr = t / 5, tc = t % 5;
#if __has_builtin(__builtin_amdgcn_wmma_f32_16x16x4_f32)
        v8f acc = {0.f, 0.f, 0.f, 0.f, 0.f, 0.f, 0.f, 0.f};
        const float* arow = &uA.gray[(16 * tr + m16) * GRAY_W + 16 * tc + 2 * half];
#pragma unroll
        for (int kc = 0; kc < 6; ++kc) {
            // A layout: K = 4*kc + 2*half + v  (v = vector element)
            v2f a = *(const v2f*)(arow + 4 * kc);
            // B layout: K = 4*kc + 2*half + v, N = m16
            int k0 = 4 * kc + 2 * half;
            v2f b = {bandw(k0, m16), bandw(k0 + 1, m16)};
            acc = __builtin_amdgcn_wmma_f32_16x16x4_f32(
                false, a, false, b, (short)0, acc, false, false);
        }
#pragma unroll
        for (int r = 0; r < 8; ++r)   // D: M = r + 8*half, N = m16
            hout[(16 * tr + r + 8 * half) * HOUT_W + 16 * tc + m16] = acc[r];
#else
        // VALU fallback: each lane computes 8 rows of one column of the 16x16 tile.
        for (int rr = 0; rr < 8; ++rr) {
            int r = 16 * tr + rr + 8 * half;
            int c = 16 * tc + m16;
            float s = 0.0f;
#pragma unroll
            for (int j = 0; j < 7; ++j) s += gcoef(j) * uA.gray[r * GRAY_W + c + j];
            hout[r * HOUT_W + c] = s;
        }
#endif
    }
    __syncthreads();

    // ---- Stage B2: vertical Gaussian via WMMA (A = band, B = hout data) --
    // smooth[s][c] = sum_j g1[j] * hout[s+j][c]
    for (int t = wave; t < 25; t += 8) {
        int tr = t / 5, tc = t % 5;
#if __has_builtin(__builtin_amdgcn_wmma_f32_16x16x4_f32)
        v8f acc = {0.f, 0.f, 0.f, 0.f, 0.f, 0.f, 0.f, 0.f};
#pragma unroll
        for (int kc = 0; kc < 6; ++kc) {
            int k0 = 4 * kc + 2 * half;
            v2f a = {bandw(k0, m16), bandw(k0 + 1, m16)};          // A[m][k]=g1[k-m]
            v2f b = {hout[(16 * tr + k0) * HOUT_W + 16 * tc + m16],
                     hout[(16 * tr + k0 + 1) * HOUT_W + 16 * tc + m16]};
            acc = __builtin_amdgcn_wmma_f32_16x16x4_f32(
                false, a, false, b, (short)0, acc, false, false);
        }
#pragma unroll
        for (int r = 0; r < 8; ++r)
            uA.smooth[(16 * tr + r + 8 * half) * SM_W + 16 * tc + m16] = acc[r];
#else
        for (int rr = 0; rr < 8; ++rr) {
            int r = 16 * tr + rr + 8 * half;
            int c = 16 * tc + m16;
            float s = 0.0f;
#pragma unroll
            for (int j = 0; j < 7; ++j) s += gcoef(j) * hout[(r + j) * HOUT_W + c];
            uA.smooth[r * SM_W + c] = s;
        }
#endif
    }
    __syncthreads();

    // ---- Stage C: Sobel + magnitude + orientation + histogram ------------
    // smooth local (s,c) maps to image (ty*64 - 1 + s, tx*64 - 1 + c).
    for (int i = 0; i < 16; ++i) {
        int p = i * 256 + tid;
        int oy = p >> 6, ox = p & 63;
        const float* s0 = &uA.smooth[oy * SM_W + ox];
        float a00 = s0[0],        a01 = s0[1],        a02 = s0[2];
        float a10 = s0[SM_W],                         a12 = s0[SM_W + 2];
        float a20 = s0[2 * SM_W], a21 = s0[2 * SM_W + 1], a22 = s0[2 * SM_W + 2];
        float gx = (a00 - a02) + 2.0f * (a10 - a12) + (a20 - a22);
        float gy = (a00 + 2.0f * a01 + a02) - (a20 + 2.0f * a21 + a22);
        float m  = sqrtf(gx * gx + gy * gy);

        int Y = ty * TILE + oy, X = tx * TILE + ox;
        mag[((size_t)img << 18) + (size_t)(Y * 512 + X)] = m;

        int bin = (int)(m * ((float)NBINS / HIST_MAX));
        bin = bin > NBINS - 1 ? NBINS - 1 : bin;
        atomicAdd(&hist[bin], 1u);

        if (img == 0) {
            // grad_ori = round((atan2*180/3.14159 + 180)/45); idx = ori % 8
            float ang = atan2f(gy, gx) * (180.0f / 3.14159f) + 180.0f;
            int q = (int)rintf(ang * (1.0f / 45.0f));
            idxMap[Y * 512 + X] = (unsigned char)(q & 7);
        }
    }
    __syncthreads();

    // Flush LDS histogram to global (order-independent -> deterministic).
    for (int i = tid; i < NBINS; i += 256) {
        unsigned c = hist[i];
        if (c) atomicAdd(&gHist[img * NBINS + i], c);
    }
}

// ---- Kernel 2: per-image 0.85 quantile from histogram ---------------------
__global__ __launch_bounds__(256) void k_thresh(const unsigned* __restrict__ gHist,
                                                float* __restrict__ thr) {
    __shared__ unsigned chunk[256];
    __shared__ float sThr;
    const int img = blockIdx.x, tid = threadIdx.x;
    const unsigned* h = gHist + img * NBINS;
    const int per = NBINS / 256;

    unsigned s = 0;
    for (int j = 0; j < per; ++j) s += h[tid * per + j];
    chunk[tid] = s;
    __syncthreads();
    if (tid == 0) {
        unsigned run = 0;
        for (int i = 0; i < 256; ++i) { unsigned c = chunk[i]; chunk[i] = run; run += c; }
        sThr = HIST_MAX;  // fallback
    }
    __syncthreads();

    const float target = 0.85f * (float)(512 * 512 - 1);   // rank 222821.55
    unsigned cum = chunk[tid];
    for (int j = 0; j < per; ++j) {
        int b = tid * per + j;
        unsigned c = h[b];
        if (c && (float)cum <= target && target < (float)(cum + c)) {
            float frac = (target - (float)cum) / (float)c;   // linear interp in bin
            sThr = ((float)b + frac) * (HIST_MAX / (float)NBINS);
        }
        cum += c;
    }
    __syncthreads();
    if (tid == 0) thr[img] = sThr;
}

// ---- Kernel 3: NMS + threshold --------------------------------------------
// Direction channel k -> subtracted-neighbor offset (replicates direct_w,
// including the k==4 special case (+1,+1)). idx_neg = (k+4)&7 (reference
// quirk: uses channel index arithmetic, not geometric opposite).
__global__ __launch_bounds__(256) void k_nms(const float* __restrict__ mag,
                                             const unsigned char* __restrict__ idxMap,
                                             const float* __restrict__ thr,
                                             float* __restrict__ out) {
    const int OY[8] = {-1, -1, -1, 0, 1, 0, 1, 1};
    const int OX[8] = {-1,  0,  1, -1, 1, 1, -1, 0};

    int n = blockIdx.x * 256 + threadIdx.x;          // 0 .. 8388607
    int img = n >> 18;
    int rem = n & 262143;
    int y = rem >> 9, x = rem & 511;

    float m = mag[(size_t)n];
    int k  = idxMap[rem];                            // image-0 indices, broadcast
    int k2 = (k + 4) & 7;

    int y1 = y + OY[k],  x1 = x + OX[k];
    int y2 = y + OY[k2], x2 = x + OX[k2];
    size_t base = (size_t)img << 18;
    float nb1 = (y1 >= 0 && y1 < 512 && x1 >= 0 && x1 < 512)
                    ? mag[base + (size_t)(y1 * 512 + x1)] : 0.0f;  // SAME zero pad
    float nb2 = (y2 >= 0 && y2 < 512 && x2 >= 0 && x2 < 512)
                    ? mag[base + (size_t)(y2 * 512 + x2)] : 0.0f;
    float pos = m - nb1, neg = m - nb2;

    float mm = (fminf(pos, neg) > 0.0f) ? m : 0.0f;  // is_max mask
    out[(size_t)n] = (mm < thr[img]) ? 0.0f : mm;    // threshold
}

// ---------------------------------------------------------------------------
extern "C" void kernel_launch(void* const* d_in, const int* in_sizes, int n_in,
                              void* d_out, int out_size, void* d_ws, size_t ws_size,
                              hipStream_t stream) {
    (void)in_sizes; (void)n_in; (void)out_size; (void)ws_size;
    const float* x = (const float*)d_in[0];
    float* out = (float*)d_out;

    char* ws = (char*)d_ws;
    float*         magBuf = (float*)ws;                                 // 33,554,432 B
    unsigned char* idxMap = (unsigned char*)(ws + 33554432u);           //    262,144 B
    unsigned*      gHist  = (unsigned*)(ws + 33816576u);                //    262,144 B
    float*         thrBuf = (float*)(ws + 34078720u);                   //        128 B

    hipMemsetAsync(gHist, 0, 32u * NBINS * sizeof(unsigned), stream);   // capture-legal

    dim3 grid1(512 / TILE, 512 / TILE, 32);
    k_grad<<<grid1, 256, 0, stream>>>(x, magBuf, idxMap, gHist);
    k_thresh<<<32, 256, 0, stream>>>(gHist, thrBuf);
    k_nms<<<(32 * 512 * 512) / 256, 256, 0, stream>>>(magBuf, idxMap, thrBuf, out);
}